// GAT_19301583028500
// MI455X (gfx1250) — compile-verified
//
#include <hip/hip_runtime.h>
#include <hip/hip_bf16.h>

#define N_NODES   50000
#define IN_DIM    256
#define HF        256
#define HEADS     4
#define FDIM      64
#define NEDGE     800000
#define NEG_SLOPE 0.2f

typedef float v2f __attribute__((ext_vector_type(2)));
typedef float v8f __attribute__((ext_vector_type(8)));

// ---------------------------------------------------------------------------
// GEMM: dst[N,256] = x[N,256] @ Wmat[256,256] using V_WMMA_F32_16X16X4_F32.
// Block = 512 threads = 16 waves. Block handles a 16-row strip; wave w owns
// columns [16w, 16w+16). A-strip staged in LDS with stride 260 (pad -> no
// bank conflicts: bank = (4*row + k) mod 64 distinct for rows 0..15).
// ---------------------------------------------------------------------------
__global__ __launch_bounds__(512)
void gat_gemm_wmma(const float* __restrict__ x,
                   const float* __restrict__ Wmat,
                   float* __restrict__ dst)
{
    __shared__ float lds[16 * 260];
    const int r0 = blockIdx.x * 16;

    // cooperative load: 16 rows x 256 cols, fully coalesced (4096 / 512 = 8 each)
    for (int t = threadIdx.x; t < 16 * 256; t += 512) {
        const int row = t >> 8;
        const int col = t & 255;
        lds[row * 260 + col] = x[(size_t)(r0 + row) * IN_DIM + col];
    }
    __syncthreads();

    const int wave = threadIdx.x >> 5;
    const int lane = threadIdx.x & 31;
    const int c0   = wave * 16;          // this wave's output column tile
    const int mrow = lane & 15;          // M (for A) / N (for B,C,D) index
    const int kb   = (lane >> 4) * 2;    // lanes 16-31 hold K+2, K+3

    v8f acc = {};
    const v2f* lds2 = (const v2f*)lds;

    for (int k0 = 0; k0 < IN_DIM; k0 += 4) {
        // A 16x4: VGPR0 = K(kb), VGPR1 = K(kb+1); row = lane&15. Aligned float2.
        v2f a = lds2[(mrow * 260 + k0 + kb) >> 1];
        // B 4x16: lane&15 = output column, VGPR0/1 = rows kb / kb+1 of W
        v2f b;
        b.x = Wmat[(size_t)(k0 + kb)     * HF + c0 + mrow];
        b.y = Wmat[(size_t)(k0 + kb + 1) * HF + c0 + mrow];
        // (neg_a, A, neg_b, B, c_mod, C, reuse_a, reuse_b)
        acc = __builtin_amdgcn_wmma_f32_16x16x4_f32(
            false, a, false, b, (short)0, acc, false, false);
    }

    // D layout: VGPR j -> row j (lanes 0-15) / row j+8 (lanes 16-31)
    const int colw  = c0 + mrow;
    const int rbase = (lane >> 4) * 8;
#pragma unroll
    for (int j = 0; j < 8; ++j) {
        dst[(size_t)(r0 + rbase + j) * HF + colw] = acc[j];
    }
}

// ---------------------------------------------------------------------------
// Per-(node,head) attention scores s_src/s_trg + zero the softmax denominators.
// One thread per (n,h): 64-length dot products, float4 vectorized.
// ---------------------------------------------------------------------------
__global__ void gat_scores(const float* __restrict__ proj,
                           const float* __restrict__ a_src,
                           const float* __restrict__ a_trg,
                           float* __restrict__ s_src,
                           float* __restrict__ s_trg,
                           float* __restrict__ denom)
{
    const int idx = blockIdx.x * blockDim.x + threadIdx.x;
    if (idx >= N_NODES * HEADS) return;
    const int n = idx >> 2;
    const int h = idx & 3;
    const float4* p  = (const float4*)(proj + (size_t)n * HF + h * FDIM);
    const float4* as = (const float4*)(a_src + h * FDIM);
    const float4* at = (const float4*)(a_trg + h * FDIM);
    float ss = 0.f, st = 0.f;
#pragma unroll
    for (int i = 0; i < FDIM / 4; ++i) {
        float4 v = p[i], u = as[i], w = at[i];
        ss += v.x * u.x + v.y * u.y + v.z * u.z + v.w * u.w;
        st += v.x * w.x + v.y * w.y + v.z * w.z + v.w * w.w;
    }
    s_src[idx] = ss;
    s_trg[idx] = st;
    denom[idx] = 0.f;   // fresh every call -> deterministic replay
}

// ---------------------------------------------------------------------------
// Edge scoring: e = leaky(s_src[src]+s_trg[trg]); numerator = exp(e);
// accumulate per-target denominators. One thread per (edge, head).
// Global-max subtraction of the reference cancels in ex/denom.
// ---------------------------------------------------------------------------
__global__ void gat_edge_scores(const int* __restrict__ esrc,
                                const int* __restrict__ etrg,
                                const float* __restrict__ s_src,
                                const float* __restrict__ s_trg,
                                float* __restrict__ exbuf,
                                float* __restrict__ denom)
{
    const int idx = blockIdx.x * blockDim.x + threadIdx.x;
    if (idx >= NEDGE * HEADS) return;
    const int e = idx >> 2;
    const int h = idx & 3;
    const int s = esrc[e];
    const int t = etrg[e];
    float v = s_src[s * HEADS + h] + s_trg[t * HEADS + h];
    v = v > 0.f ? v : NEG_SLOPE * v;
    const float ex = __expf(v);
    exbuf[idx] = ex;
    atomicAdd(&denom[t * HEADS + h], ex);
}

// ---------------------------------------------------------------------------
// Aggregation: one wave (32 lanes) per edge. Lanes stride the 256 feature
// columns (8 each) -> coalesced proj reads and coalesced f32 atomics into out
// (out already holds the skip projection from the second GEMM).
// ---------------------------------------------------------------------------
__global__ __launch_bounds__(256)
void gat_aggregate(const int* __restrict__ esrc,
                   const int* __restrict__ etrg,
                   const float* __restrict__ exbuf,
                   const float* __restrict__ denom,
                   const float* __restrict__ proj,
                   float* __restrict__ out)
{
    const int e = blockIdx.x * 8 + (threadIdx.x >> 5);
    if (e >= NEDGE) return;
    const int lane = threadIdx.x & 31;
    const int s = esrc[e];
    const int t = etrg[e];
    const float4 exv = *(const float4*)(exbuf + (size_t)e * HEADS);
    const float4 dn  = *(const float4*)(denom + (size_t)t * HEADS);
    float attn[HEADS] = { exv.x / (dn.x + 1e-16f),
                          exv.y / (dn.y + 1e-16f),
                          exv.z / (dn.z + 1e-16f),
                          exv.w / (dn.w + 1e-16f) };
    const float* prow = proj + (size_t)s * HF;
    float*       orow = out  + (size_t)t * HF;
#pragma unroll
    for (int i = 0; i < 8; ++i) {
        const int col = lane + 32 * i;
        atomicAdd(&orow[col], attn[col >> 6] * prow[col]);
    }
}

// ---------------------------------------------------------------------------
// Finalize: out = leaky(out + bias), float4 vectorized, in place.
// ---------------------------------------------------------------------------
__global__ void gat_finalize(float* __restrict__ out,
                             const float* __restrict__ bias)
{
    const int idx = blockIdx.x * blockDim.x + threadIdx.x;  // over N*256/4
    if (idx >= N_NODES * (HF / 4)) return;
    float4 v = ((float4*)out)[idx];
    float4 b = ((const float4*)bias)[idx & (HF / 4 - 1)];
    v.x += b.x; v.y += b.y; v.z += b.z; v.w += b.w;
    v.x = v.x > 0.f ? v.x : NEG_SLOPE * v.x;
    v.y = v.y > 0.f ? v.y : NEG_SLOPE * v.y;
    v.z = v.z > 0.f ? v.z : NEG_SLOPE * v.z;
    v.w = v.w > 0.f ? v.w : NEG_SLOPE * v.w;
    ((float4*)out)[idx] = v;
}

extern "C" void kernel_launch(void* const* d_in, const int* in_sizes, int n_in,
                              void* d_out, int out_size, void* d_ws, size_t ws_size,
                              hipStream_t stream)
{
    const float* x      = (const float*)d_in[0];   // [N, 256]
    const float* W      = (const float*)d_in[1];   // [256, 256]
    const float* a_src  = (const float*)d_in[2];   // [4, 64]
    const float* a_trg  = (const float*)d_in[3];   // [4, 64]
    const float* skip_w = (const float*)d_in[4];   // [256, 256]
    const float* bias   = (const float*)d_in[5];   // [256]
    const int*   esrc   = (const int*)d_in[6];     // [E]
    const int*   etrg   = (const int*)d_in[7];     // [E]
    float* out = (float*)d_out;                    // [N, 256]

    // workspace layout (floats)
    float* ws     = (float*)d_ws;
    float* proj   = ws;                                   // N*256   = 12,800,000
    float* s_src  = proj  + (size_t)N_NODES * HF;         // N*4     =    200,000
    float* s_trg  = s_src + (size_t)N_NODES * HEADS;      // N*4
    float* denom  = s_trg + (size_t)N_NODES * HEADS;      // N*4
    float* exbuf  = denom + (size_t)N_NODES * HEADS;      // E*4     =  3,200,000

    // 1) proj = x @ W   (WMMA f32)
    gat_gemm_wmma<<<dim3(N_NODES / 16), 512, 0, stream>>>(x, W, proj);
    // 2) out = x @ skip_w  (WMMA f32, full overwrite of d_out)
    gat_gemm_wmma<<<dim3(N_NODES / 16), 512, 0, stream>>>(x, skip_w, out);
    // 3) per-(node,head) scores + denom := 0
    gat_scores<<<(N_NODES * HEADS + 255) / 256, 256, 0, stream>>>(
        proj, a_src, a_trg, s_src, s_trg, denom);
    // 4) edge numerators + denominator scatter-add
    gat_edge_scores<<<(NEDGE * HEADS + 255) / 256, 256, 0, stream>>>(
        esrc, etrg, s_src, s_trg, exbuf, denom);
    // 5) attention-weighted aggregation into out
    gat_aggregate<<<(NEDGE + 7) / 8, 256, 0, stream>>>(
        esrc, etrg, exbuf, denom, proj, out);
    // 6) bias + output LeakyReLU
    gat_finalize<<<(N_NODES * (HF / 4) + 255) / 256, 256, 0, stream>>>(out, bias);
}